// SimpleSSM_15530601742691
// MI455X (gfx1250) — compile-verified
//
#include <hip/hip_runtime.h>
#include <hip/hip_bf16.h>

// Problem constants (SimpleSSM)
#define BZ    8
#define DI    256
#define DS    512
#define DO    256
#define LSEQ  4096
#define TCH   256                  // scan chunk length
#define NCHUNK (LSEQ / TCH)        // 16

// LDS padded strides (floats): chosen so WMMA fragment reads are conflict-free
#define UP    68                   // u tile row stride (row = one i, 64 l cols)
#define XP    516                  // X tile row stride (row = one l, 512 s cols)

typedef float v2f __attribute__((ext_vector_type(2)));
typedef float v4f __attribute__((ext_vector_type(4)));
typedef float v8f __attribute__((ext_vector_type(8)));

static __device__ __forceinline__ v8f wmma_f32_16x16x4(v2f a, v2f b, v8f c) {
    // V_WMMA_F32_16X16X4_F32 : D = A(16x4) * B(4x16) + C(16x16), full fp32
    return __builtin_amdgcn_wmma_f32_16x16x4_f32(
        /*neg_a=*/false, a, /*neg_b=*/false, b,
        /*c_mod=*/(short)0, c, /*reuse_a=*/false, /*reuse_b=*/false);
}

// ---------------------------------------------------------------------------
// Kernel 0: A_diag[s] = -softplus(A_unconstrained[s])   (512 threads)
// ---------------------------------------------------------------------------
__global__ void k_adiag(const float* __restrict__ a_un, float* __restrict__ adiag) {
    int s = threadIdx.x;
    float x  = a_un[s];
    float sp = fmaxf(x, 0.0f) + log1pf(expf(-fabsf(x)));   // stable softplus
    adiag[s] = -sp;
}

// ---------------------------------------------------------------------------
// Kernel 1: GEMM1  X[b,l,s] (pre-scan) = sum_i B[s,i] * u[b,i,l]
//   grid = BZ * (LSEQ/64) = 512 blocks, 256 threads (8 waves).
//   block tile: 512(s) x 64(l). The u tile (256 x 64) is shared by all 8
//   waves -> staged once into LDS (68-float padded rows), then WMMA
//   B-fragments come from LDS. B_mat rows are per-wave-private -> global/L2.
// ---------------------------------------------------------------------------
__global__ __launch_bounds__(256) void k_gemm1(const float* __restrict__ u,
                                               const float* __restrict__ Bm,
                                               float* __restrict__ X) {
    extern __shared__ char smem_raw[];
    float* s_u = (float*)smem_raw;              // [DI][UP] = 256*68 floats

    const int b      = blockIdx.x >> 6;         // [0,8)
    const int l0base = (blockIdx.x & 63) * 64;  // [0,4096) step 64
    const int t      = threadIdx.x;
    const int wave   = t >> 5;
    const int lane   = t & 31;
    const int mrow   = lane & 15;               // A: M row
    const int khalf  = (lane >> 4) << 1;        // A/B: K sub-offset {0,2}
    const int ncol   = lane & 15;               // B/D: N column
    const int hi     = (lane >> 4) << 3;        // C/D: +8 row offset lanes 16..31

    const float* __restrict__ ub = u + (size_t)b * DI * LSEQ;

    // Hint the B_mat rows this wave is about to stream (global_prefetch_b8)
    __builtin_prefetch(Bm + (size_t)(wave * 64 + mrow) * DI, 0, 3);

    // ---- stage u tile: thread t copies row i=t (64 floats, 16x float4) ----
    #pragma unroll
    for (int c = 0; c < 16; ++c) {
        const float4 v = *(const float4*)(ub + (size_t)t * LSEQ + l0base + c * 4);
        *(float4*)(&s_u[t * UP + c * 4]) = v;
    }
    __syncthreads();

    const v8f vzero = {0.f,0.f,0.f,0.f,0.f,0.f,0.f,0.f};

    for (int half = 0; half < 2; ++half) {
        const int st0 = wave * 4 + half * 2;    // two s-tile rows
        v8f acc[2][4];
        #pragma unroll
        for (int i = 0; i < 2; ++i)
            #pragma unroll
            for (int j = 0; j < 4; ++j) acc[i][j] = vzero;

        for (int k0 = 0; k0 < DI; k0 += 4) {
            v2f afrag[2];
            #pragma unroll
            for (int i = 0; i < 2; ++i) {
                const int srow = (st0 + i) * 16 + mrow;
                afrag[i] = *(const v2f*)(Bm + (size_t)srow * DI + k0 + khalf);
            }
            v2f bfrag[4];
            #pragma unroll
            for (int j = 0; j < 4; ++j) {
                const int lcol = j * 16 + ncol;
                bfrag[j].x = s_u[(k0 + khalf)     * UP + lcol];
                bfrag[j].y = s_u[(k0 + khalf + 1) * UP + lcol];
            }
            #pragma unroll
            for (int i = 0; i < 2; ++i)
                #pragma unroll
                for (int j = 0; j < 4; ++j)
                    acc[i][j] = wmma_f32_16x16x4(afrag[i], bfrag[j], acc[i][j]);
        }

        // Store D tiles: lane covers s = st*16 + hi + [0,8) contiguous -> 2x B128
        #pragma unroll
        for (int i = 0; i < 2; ++i) {
            const int sbase = (st0 + i) * 16 + hi;
            #pragma unroll
            for (int j = 0; j < 4; ++j) {
                const int l = l0base + j * 16 + ncol;
                float* xp = X + ((size_t)b * LSEQ + l) * DS + sbase;
                v4f lo  = __builtin_shufflevector(acc[i][j], acc[i][j], 0,1,2,3);
                v4f hiv = __builtin_shufflevector(acc[i][j], acc[i][j], 4,5,6,7);
                *(v4f*)(xp)     = lo;
                *(v4f*)(xp + 4) = hiv;
            }
        }
    }
}

// ---------------------------------------------------------------------------
// Kernel 2: per-chunk local scan (carry-in = 0), in place on X.
//   grid = BZ*NCHUNK = 128 blocks, 256 threads; thread owns s and s+256.
//   Coalesced: lanes hit consecutive s within the (b,l,s) layout.
// ---------------------------------------------------------------------------
__global__ __launch_bounds__(256) void k_scanlocal(const float* __restrict__ adiag,
                                                   float* __restrict__ X,
                                                   float* __restrict__ carry_local) {
    const int b = blockIdx.x >> 4, c = blockIdx.x & 15;
    const int s1 = threadIdx.x, s2 = threadIdx.x + 256;
    const float a1 = adiag[s1], a2 = adiag[s2];
    float x1 = 0.f, x2 = 0.f;
    const size_t base = ((size_t)b * LSEQ + (size_t)c * TCH) * DS;
    for (int tt = 0; tt < TCH; ++tt) {
        const size_t idx = base + (size_t)tt * DS;
        x1 = fmaf(a1, x1, X[idx + s1]);
        x2 = fmaf(a2, x2, X[idx + s2]);
        X[idx + s1] = x1;
        X[idx + s2] = x2;
    }
    const size_t ci = ((size_t)b * NCHUNK + c) * DS;
    carry_local[ci + s1] = x1;
    carry_local[ci + s2] = x2;
}

// ---------------------------------------------------------------------------
// Kernel 3: serial carry combine across the 16 chunks (tiny).
//   cin_0 = h0;  cin_{c+1} = a^TCH * cin_c + local_final_c
// ---------------------------------------------------------------------------
__global__ __launch_bounds__(256) void k_combine(const float* __restrict__ adiag,
                                                 const float* __restrict__ h0,
                                                 const float* __restrict__ carry_local,
                                                 float* __restrict__ carryin) {
    const int b = blockIdx.x;
    #pragma unroll
    for (int rep = 0; rep < 2; ++rep) {
        const int s = threadIdx.x + rep * 256;
        const float a = adiag[s];
        float aT = a;
        #pragma unroll
        for (int i = 0; i < 8; ++i) aT = aT * aT;   // a^(2^8) = a^256 = a^TCH
        float cin = h0[s];
        for (int c = 0; c < NCHUNK; ++c) {
            const size_t idx = ((size_t)b * NCHUNK + c) * DS + s;
            carryin[idx] = cin;
            cin = fmaf(aT, cin, carry_local[idx]);
        }
    }
}

// ---------------------------------------------------------------------------
// Kernel 4: fixup  X[b, t0+t, s] += a^(t+1) * carry_in   (grid = 128 blocks)
// ---------------------------------------------------------------------------
__global__ __launch_bounds__(256) void k_fixup(const float* __restrict__ adiag,
                                               const float* __restrict__ carryin,
                                               float* __restrict__ X) {
    const int b = blockIdx.x >> 4, c = blockIdx.x & 15;
    const int s1 = threadIdx.x, s2 = threadIdx.x + 256;
    const float a1 = adiag[s1], a2 = adiag[s2];
    const size_t ci = ((size_t)b * NCHUNK + c) * DS;
    const float c1 = carryin[ci + s1], c2 = carryin[ci + s2];
    float p1 = a1, p2 = a2;
    const size_t base = ((size_t)b * LSEQ + (size_t)c * TCH) * DS;
    for (int tt = 0; tt < TCH; ++tt) {
        const size_t idx = base + (size_t)tt * DS;
        X[idx + s1] = fmaf(p1, c1, X[idx + s1]);  p1 *= a1;
        X[idx + s2] = fmaf(p2, c2, X[idx + s2]);  p2 *= a2;
    }
}

// ---------------------------------------------------------------------------
// Kernel 5: GEMM2  Y[b,o,l] = sum_s C[o,s]*X[b,l,s] + sum_i D[o,i]*u[b,i,l]
//   grid = BZ * (LSEQ/64) = 512 blocks; wave owns 2 o-tiles x 4 l-tiles.
//   The X tile (64 x 512) and u tile (256 x 64) are shared by all waves:
//   staged into LDS sequentially (same buffer), WMMA B-fragments from LDS.
// ---------------------------------------------------------------------------
__global__ __launch_bounds__(256) void k_gemm2(const float* __restrict__ X,
                                               const float* __restrict__ u,
                                               const float* __restrict__ Cm,
                                               const float* __restrict__ Dm,
                                               float* __restrict__ Y) {
    extern __shared__ char smem_raw[];
    float* s_x = (float*)smem_raw;              // phase 1: [64][XP]
    float* s_u = (float*)smem_raw;              // phase 2: [DI][UP] (reused)

    const int b      = blockIdx.x >> 6;
    const int l0base = (blockIdx.x & 63) * 64;
    const int t      = threadIdx.x;
    const int wave   = t >> 5;
    const int lane   = t & 31;
    const int mrow   = lane & 15;
    const int khalf  = (lane >> 4) << 1;
    const int ncol   = lane & 15;
    const int hi     = (lane >> 4) << 3;

    const float* __restrict__ ub = u + (size_t)b * DI * LSEQ;
    const float* __restrict__ Xb = X + (size_t)b * LSEQ * DS;

    __builtin_prefetch(Cm + (size_t)(wave * 32 + mrow) * DS, 0, 3);

    // ---- stage X tile: 64 rows x 512 cols; 4 threads per row, float4 ----
    {
        const int lr = t >> 2;                  // [0,64)
        const int cb = (t & 3) * 128;           // column block
        #pragma unroll
        for (int c = 0; c < 32; ++c) {
            const float4 v = *(const float4*)(Xb + (size_t)(l0base + lr) * DS + cb + c * 4);
            *(float4*)(&s_x[lr * XP + cb + c * 4]) = v;
        }
    }
    __syncthreads();

    const v8f vzero = {0.f,0.f,0.f,0.f,0.f,0.f,0.f,0.f};
    v8f acc[2][4];
    #pragma unroll
    for (int i = 0; i < 2; ++i)
        #pragma unroll
        for (int j = 0; j < 4; ++j) acc[i][j] = vzero;

    // K-loop 1: C (256x512) x X^T — A: contiguous v2f from global/L2,
    //                               B: contiguous v2f from LDS (stride XP)
    for (int k0 = 0; k0 < DS; k0 += 4) {
        v2f afrag[2];
        #pragma unroll
        for (int i = 0; i < 2; ++i) {
            const int orow = (wave * 2 + i) * 16 + mrow;
            afrag[i] = *(const v2f*)(Cm + (size_t)orow * DS + k0 + khalf);
        }
        v2f bfrag[4];
        #pragma unroll
        for (int j = 0; j < 4; ++j) {
            const int lr = j * 16 + ncol;
            bfrag[j] = *(const v2f*)(&s_x[lr * XP + k0 + khalf]);
        }
        #pragma unroll
        for (int i = 0; i < 2; ++i)
            #pragma unroll
            for (int j = 0; j < 4; ++j)
                acc[i][j] = wmma_f32_16x16x4(afrag[i], bfrag[j], acc[i][j]);
    }

    // ---- re-stage LDS with the u tile (all waves must be done with X) ----
    __syncthreads();
    #pragma unroll
    for (int c = 0; c < 16; ++c) {
        const float4 v = *(const float4*)(ub + (size_t)t * LSEQ + l0base + c * 4);
        *(float4*)(&s_u[t * UP + c * 4]) = v;
    }
    __syncthreads();

    // K-loop 2: D (256x256) x u  (D==0 numerically, kept for generality)
    for (int k0 = 0; k0 < DI; k0 += 4) {
        v2f afrag[2];
        #pragma unroll
        for (int i = 0; i < 2; ++i) {
            const int orow = (wave * 2 + i) * 16 + mrow;
            afrag[i] = *(const v2f*)(Dm + (size_t)orow * DI + k0 + khalf);
        }
        v2f bfrag[4];
        #pragma unroll
        for (int j = 0; j < 4; ++j) {
            const int lcol = j * 16 + ncol;
            bfrag[j].x = s_u[(k0 + khalf)     * UP + lcol];
            bfrag[j].y = s_u[(k0 + khalf + 1) * UP + lcol];
        }
        #pragma unroll
        for (int i = 0; i < 2; ++i)
            #pragma unroll
            for (int j = 0; j < 4; ++j)
                acc[i][j] = wmma_f32_16x16x4(afrag[i], bfrag[j], acc[i][j]);
    }

    // Store Y[b,o,l]; rows scattered by v, columns contiguous across lanes 0..15
    #pragma unroll
    for (int i = 0; i < 2; ++i) {
        const int obase = (wave * 2 + i) * 16 + hi;
        #pragma unroll
        for (int j = 0; j < 4; ++j) {
            const int l = l0base + j * 16 + ncol;
            #pragma unroll
            for (int v = 0; v < 8; ++v)
                Y[((size_t)b * DO + obase + v) * LSEQ + l] = acc[i][j][v];
        }
    }
}

// ---------------------------------------------------------------------------
// Host-side launch
// ---------------------------------------------------------------------------
extern "C" void kernel_launch(void* const* d_in, const int* in_sizes, int n_in,
                              void* d_out, int out_size, void* d_ws, size_t ws_size,
                              hipStream_t stream) {
    (void)in_sizes; (void)n_in; (void)out_size; (void)ws_size;

    const float* u    = (const float*)d_in[0];  // (8,256,4096)
    const float* a_un = (const float*)d_in[1];  // (512,)
    const float* Bm   = (const float*)d_in[2];  // (512,256)
    const float* Cm   = (const float*)d_in[3];  // (256,512)
    const float* Dm   = (const float*)d_in[4];  // (256,256)
    const float* h0   = (const float*)d_in[5];  // (512,)
    float*       Y    = (float*)d_out;          // (8,256,4096)

    char* ws = (char*)d_ws;
    const size_t X_BYTES = (size_t)BZ * LSEQ * DS * sizeof(float);     // 64 MiB
    const size_t C_BYTES = (size_t)BZ * NCHUNK * DS * sizeof(float);   // 256 KiB
    float* X           = (float*)(ws);
    float* carry_local = (float*)(ws + X_BYTES);
    float* carryin     = (float*)(ws + X_BYTES + C_BYTES);
    float* adiag       = (float*)(ws + X_BYTES + 2 * C_BYTES);

    const size_t SH1 = (size_t)DI * UP * sizeof(float);   // 69,632 B  u tile
    const size_t SH2 = (size_t)64 * XP * sizeof(float);   // 132,096 B X tile

    k_adiag    <<<1,                512, 0,   stream>>>(a_un, adiag);
    k_gemm1    <<<BZ * (LSEQ / 64), 256, SH1, stream>>>(u, Bm, X);
    k_scanlocal<<<BZ * NCHUNK,      256, 0,   stream>>>(adiag, X, carry_local);
    k_combine  <<<BZ,               256, 0,   stream>>>(adiag, h0, carry_local, carryin);
    k_fixup    <<<BZ * NCHUNK,      256, 0,   stream>>>(adiag, carryin, X);
    k_gemm2    <<<BZ * (LSEQ / 64), 256, SH2, stream>>>(X, u, Cm, Dm, Y);
}